// CRF_12627203850960
// MI455X (gfx1250) — compile-verified
//
#include <hip/hip_runtime.h>
#include <hip/hip_bf16.h>
#include <stdint.h>

#define K 512
#define TMAX 1024
#define B_LAST 63
#define START_TAG 510
#define STOP_TAG 511
#define NEG_INF_V (-10000.0f)

typedef unsigned int v4u __attribute__((ext_vector_type(4)));
typedef int          v8i __attribute__((ext_vector_type(8)));
typedef int          v4i __attribute__((ext_vector_type(4)));

// One-time transpose of transitions (1 MB) so the hot loop's loads are
// coalesced: transT[i*K + j] = trans[j*K + i].
__global__ void crf_transpose_kernel(const float* __restrict__ in,
                                     float* __restrict__ outT) {
    int idx = blockIdx.x * blockDim.x + threadIdx.x;
    int i = idx & (K - 1);
    int j = idx >> 9;
    if (j < K) outT[(size_t)i * K + j] = in[(size_t)j * K + i];
}

// Tensor Data Mover: DMA one contiguous row of `nelem` f32 from global to LDS.
// D# per CDNA5 ISA: group0 = control/lds/global addr,
// group1 = dims/strides, groups 2-3 unused for 2-D tiles (tile_dim2 = 0).
__device__ __forceinline__ void tdm_load_row_f32(unsigned lds_off,
                                                 const float* gptr,
                                                 unsigned nelem) {
    unsigned long long ga = (unsigned long long)(size_t)gptr;
    v4u g0;
    g0.x = 1u;                                               // count=1, user desc
    g0.y = lds_off;                                          // lds_addr (bytes)
    g0.z = (unsigned)(ga & 0xffffffffu);                     // global_addr[31:0]
    g0.w = (unsigned)((ga >> 32) & 0x01ffffffu) | (2u << 30);// ga[56:32] | type=2
    v8i g1;
    g1[0] = 0x00020000;                          // wg_mask=0, data_size=2 (4 B)
    g1[1] = (int)(nelem << 16);                  // tensor_dim0[15:0] @ bits[63:48]
    g1[2] = (int)((nelem >> 16) | (1u << 16));   // tdim0[31:16] | tensor_dim1=1
    g1[3] = (int)(nelem << 16);                  // tdim1[31:16]=0 | tile_dim0
    g1[4] = 1;                                   // tile_dim1=1, tile_dim2=0
    g1[5] = (int)nelem;                          // tensor_dim0_stride[31:0]
    g1[6] = 0;                                   // stride0 hi | stride1 lo
    g1[7] = 0;                                   // stride1 hi
    v4i z4 = {0, 0, 0, 0};
#if defined(__clang_major__) && __clang_major__ >= 23
    v8i z8 = {0, 0, 0, 0, 0, 0, 0, 0};
    __builtin_amdgcn_tensor_load_to_lds(g0, g1, z4, z4, z8, 0);
#else
    __builtin_amdgcn_tensor_load_to_lds(g0, g1, z4, z4, 0);
#endif
}

// Single-workgroup Viterbi: 1024 threads = 32 wave32 waves on one WGP.
// Thread tid handles output j = tid&511, i-range [sub*256, sub*256+256).
__global__ __launch_bounds__(1024) void crf_viterbi_kernel(
    const float* __restrict__ feats,     // [64,1024,512], only batch 63 used
    const float* __restrict__ trans,     // [512,512] original (for STOP row)
    const float* __restrict__ transT,    // [512,512] transposed (ws)
    const long long* __restrict__ lengths,
    unsigned short* __restrict__ bptr,   // [1024,512] (ws)
    float* __restrict__ out)             // [1025] = {score, path[1024]}
{
    __shared__ float fv[K];
    __shared__ float featbuf[2][K];
    __shared__ float stoprow[K];
    __shared__ float pmax[1024];
    __shared__ int   parg[1024];

    const int tid = threadIdx.x;
    const int j   = tid & (K - 1);
    const int sub = tid >> 9;                    // 0 or 1
    const int len = (int)lengths[B_LAST];
    const float* featB = feats + (size_t)B_LAST * TMAX * K;

    if (tid < K) fv[tid] = (tid == START_TAG) ? 0.0f : NEG_INF_V;
    if (tid >= len) out[1 + tid] = -1.0f;        // padded path slots

    // Async-to-LDS (ASYNCcnt path): stage trans[STOP_TAG] row for the
    // terminal stage; stays in flight across the forward loop.
    if (tid < K) {
        unsigned lva  = (unsigned)(size_t)&stoprow[tid];
        unsigned voff = (unsigned)(tid * 4);
        unsigned long long sbase =
            (unsigned long long)(size_t)(trans + (size_t)STOP_TAG * K);
        asm volatile("global_load_async_to_lds_b32 %0, %1, %2"
                     :: "v"(lva), "v"(voff), "s"(sbase) : "memory");
    }

    // TDM (TENSORcnt path): DMA feats row 0 into featbuf[0]; wave 0 only
    // (TDM ignores EXEC, so restricting to one wave avoids duplicate DMAs).
    if (tid < 32) {
        tdm_load_row_f32((unsigned)(size_t)&featbuf[0][0], featB, K);
    }

    for (int t = 0; t < len; ++t) {
        const int buf = t & 1;
        __builtin_amdgcn_s_wait_tensorcnt(0);   // feat row t landed (wave 0)
        __syncthreads();                        // fv(t) + featbuf visible to all

        // Double-buffered TDM prefetch of feat row t+1.
        if ((t + 1 < len) && (tid < 32)) {
            tdm_load_row_f32((unsigned)(size_t)&featbuf[buf ^ 1][0],
                             featB + (size_t)(t + 1) * K, K);
        }

        // Max-plus partial reduction over this thread's i-range.
        float best = -3.0e38f;
        int   arg  = sub * 256;
        const float* tb = transT + ((size_t)(sub * 256)) * K + j;
        #pragma unroll 8
        for (int i = 0; i < 256; ++i) {
            float s = fv[sub * 256 + i] + tb[(size_t)i * K]; // LDS bcast + coalesced L2
            if (s > best) { best = s; arg = sub * 256 + i; } // first-occurrence argmax
        }
        pmax[tid] = best; parg[tid] = arg;
        __syncthreads();

        if (tid < K) {
            float b0 = pmax[tid], b1 = pmax[tid + K];
            int   a0 = parg[tid], a1 = parg[tid + K];
            bool takeb = (b1 > b0);              // keep lower index on ties
            float m = takeb ? b1 : b0;
            int   a = takeb ? a1 : a0;
            bptr[(size_t)t * K + tid] = (unsigned short)a;
            fv[tid] = m + featbuf[buf][tid];     // safe: readers passed barrier
        }
        __syncthreads();
    }

    // Terminal: fv + trans[STOP_TAG] (from LDS), argmax tree (ties keep the lower slot).
    asm volatile("s_wait_asynccnt 0" ::: "memory");
    __syncthreads();
    if (tid < K) {
        pmax[tid] = fv[tid] + stoprow[tid];
        parg[tid] = tid;
    }
    __syncthreads();
    for (int s = 256; s >= 1; s >>= 1) {
        if (tid < s) {
            if (pmax[tid + s] > pmax[tid]) {
                pmax[tid] = pmax[tid + s];
                parg[tid] = parg[tid + s];
            }
        }
        __syncthreads();
    }

    if (tid == 0) {
        out[0] = pmax[0];
        int tag = parg[0];
        for (int t = len - 1; t >= 0; --t) {
            out[1 + t] = (float)tag;
            tag = (int)bptr[(size_t)t * K + tag];
        }
    }
}

extern "C" void kernel_launch(void* const* d_in, const int* in_sizes, int n_in,
                              void* d_out, int out_size, void* d_ws, size_t ws_size,
                              hipStream_t stream) {
    const float*     feats   = (const float*)d_in[0];
    const float*     trans   = (const float*)d_in[1];
    const long long* lengths = (const long long*)d_in[2];
    float*           out     = (float*)d_out;

    float*          transT = (float*)d_ws;                                       // 1 MB
    unsigned short* bptr   = (unsigned short*)((char*)d_ws + (size_t)K * K * 4); // 1 MB

    crf_transpose_kernel<<<(K * K) / 256, 256, 0, stream>>>(trans, transT);
    crf_viterbi_kernel<<<1, 1024, 0, stream>>>(feats, trans, transT, lengths, bptr, out);
}